// Model_35536559407348
// MI455X (gfx1250) — compile-verified
//
#include <hip/hip_runtime.h>
#include <hip/hip_bf16.h>

#define B_ 256
#define S_ 512
#define H_ 64
#define XS_STRIDE 65   // pad: bank-conflict-free strided LDS access
#define W_STRIDE  65
#define NTHREADS  512  // 16 wave32s per block (one batch row per block)

typedef float v2f __attribute__((ext_vector_type(2)));
typedef float v8f __attribute__((ext_vector_type(8)));

// One block = one batch row b. x_pad[b] (512x64 f32) lives in LDS for the whole
// 512-step scan; scores GEMV runs on the matrix pipe via V_WMMA_F32_16X16X4_F32
// (exact fp32) with h replicated across the 16 B-columns.
__global__ __launch_bounds__(NTHREADS) void rnn_attn_scan(
    const int* __restrict__ tokens, const int* __restrict__ lengths,
    const float* __restrict__ emb,
    const float* __restrict__ Wih, const float* __restrict__ Whh,
    const float* __restrict__ bih, const float* __restrict__ bhh,
    float* __restrict__ out)
{
  extern __shared__ float smem[];
  float* xs   = smem;                      // 512*65
  float* wih  = xs   + S_ * XS_STRIDE;     // 64*65
  float* whh  = wih  + H_ * W_STRIDE;      // 64*65
  float* sc   = whh  + H_ * W_STRIDE;      // 512 raw scores
  float* part = sc   + S_;                 // 512 partial sums (ctx / context0)
  float* wn   = part + S_;                 // 512 normalized softmax weights
  float* hvec = wn   + S_;                 // 64
  float* ctxv = hvec + H_;                 // 64
  float* hid  = ctxv + H_;                 // 64
  float* bias = hid  + H_;                 // 64 (b_ih + b_hh)
  float* wred = bias + H_;                 // 32 (per-wave max / sum)

  const int b    = blockIdx.x;
  const int tid  = threadIdx.x;
  const int lane = tid & 31;
  const int wid  = tid >> 5;
  const int row  = lane & 15;              // M index within 16-row tile
  const int half = lane >> 4;              // K-half select (ISA A-layout)
  const int len  = lengths[b];
  const float flen = (float)len;

  // ---- preload weights into LDS (padded stride 65) ----
  for (int idx = tid; idx < H_ * H_; idx += NTHREADS) {
    int j = idx >> 6, k = idx & 63;
    wih[j * W_STRIDE + k] = Wih[idx];
    whh[j * W_STRIDE + k] = Whh[idx];
  }
  if (tid < H_) { bias[tid] = bih[tid] + bhh[tid]; hid[tid] = 0.f; }

  // ---- embedding gather + masking + context0 = sum(x)/len ----
  {
    const int hh = tid & 63;               // feature index
    const int st = tid >> 6;               // s-stripe 0..7 (64 s each)
    float acc = 0.f;
    for (int s = st * 64; s < st * 64 + 64; ++s) {
      int tok = tokens[b * S_ + s];        // broadcast within 64-thread group
      float v = (s < len) ? emb[tok * H_ + hh] : 0.f;  // coalesced 256B row
      xs[s * XS_STRIDE + hh] = v;
      acc += v;
    }
    part[tid] = acc;
  }
  __syncthreads();
  if (tid < H_) {
    float s0 = 0.f;
    for (int st = 0; st < 8; ++st) s0 += part[st * 64 + tid];
    ctxv[tid] = s0 / flen;
  }
  __syncthreads();

  // ---- 512-step sequential scan, fully block-local ----
  for (int i = 0; i < S_; ++i) {
    // (a) RNN cell: h = tanh(ctx@Wihᵀ + hid@Whhᵀ + b). 64 lanes, exact fp32.
    if (tid < H_) {
      float acc = bias[tid];
      const float* wr1 = &wih[tid * W_STRIDE];  // (j+k)%64 banks: conflict-free
      const float* wr2 = &whh[tid * W_STRIDE];
      #pragma unroll 8
      for (int k = 0; k < H_; ++k)
        acc += ctxv[k] * wr1[k] + hid[k] * wr2[k];
      hvec[tid] = tanhf(acc);
    }
    __syncthreads();

    // (b) scores[s] = x[s,:]·h via WMMA f32 16x16x4, h replicated over N.
    // B operand per k-tile t: V0=h[4t+2*half], V1=h[4t+2*half+1] (ISA layout).
    v2f bmat[16];
    #pragma unroll
    for (int t = 0; t < 16; ++t) {
      bmat[t].x = hvec[4 * t + 2 * half];
      bmat[t].y = hvec[4 * t + 2 * half + 1];
    }
    #pragma unroll
    for (int c = 0; c < 2; ++c) {
      const int s0 = wid * 32 + c * 16;    // this wave's 16-row s-chunk
      const float* xrow = &xs[(s0 + row) * XS_STRIDE + 2 * half];
      v8f acc = {0.f, 0.f, 0.f, 0.f, 0.f, 0.f, 0.f, 0.f};
      #pragma unroll
      for (int t = 0; t < 16; ++t) {       // K=64 as 16 chained k-tiles
        v2f a;                             // A: V0=x[M,4t+2h], V1=x[M,4t+2h+1]
        a.x = xrow[4 * t];
        a.y = xrow[4 * t + 1];
        acc = __builtin_amdgcn_wmma_f32_16x16x4_f32(
            false, a, false, bmat[t], (short)0, acc, false, false);
      }
      // D layout: lane0 carries M=0..7 (N=0), lane16 carries M=8..15.
      if (lane == 0 || lane == 16) {
        #pragma unroll
        for (int r = 0; r < 8; ++r) sc[s0 + half * 8 + r] = acc[r];
      }
    }
    __syncthreads();

    // (c) log-softmax over all 512 positions (pads included, as reference)
    float v = sc[tid];
    float m = v;
    for (int off = 16; off; off >>= 1) m = fmaxf(m, __shfl_xor(m, off, 32));
    if (lane == 0) wred[wid] = m;
    __syncthreads();
    m = wred[0];
    for (int w = 1; w < 16; ++w) m = fmaxf(m, wred[w]);   // fixed order
    float p = expf(v - m);
    float ssum = p;
    for (int off = 16; off; off >>= 1) ssum += __shfl_xor(ssum, off, 32);
    if (lane == 0) wred[16 + wid] = ssum;
    __syncthreads();
    float tot = 0.f;
    for (int w = 0; w < 16; ++w) tot += wred[16 + w];     // fixed order
    float lw = (v - m) - logf(tot);
    out[((size_t)b * S_ + (size_t)i) * S_ + tid] = lw;    // coalesced 2KB row
    wn[tid] = p / tot;
    __syncthreads();

    // (d) ctx_new[h] = sum_s wn[s]*x[s,h]; stripes then deterministic fold
    {
      const int hh = tid & 63;
      const int st = tid >> 6;
      float acc2 = 0.f;
      for (int s = st * 64; s < st * 64 + 64; ++s)
        acc2 += wn[s] * xs[s * XS_STRIDE + hh];           // conflict-free
      part[tid] = acc2;
    }
    __syncthreads();

    // (e) carry update only while i < len (reference `active` mask)
    if (tid < H_) {
      float cnew = 0.f;
      for (int st = 0; st < 8; ++st) cnew += part[st * 64 + tid];
      if (i < len) { ctxv[tid] = cnew; hid[tid] = hvec[tid]; }
    }
    __syncthreads();
  }
}

extern "C" void kernel_launch(void* const* d_in, const int* in_sizes, int n_in,
                              void* d_out, int out_size, void* d_ws, size_t ws_size,
                              hipStream_t stream) {
  const int*   tokens  = (const int*)d_in[0];
  const int*   lengths = (const int*)d_in[1];
  const float* emb     = (const float*)d_in[2];
  const float* Wih     = (const float*)d_in[3];
  const float* Whh     = (const float*)d_in[4];
  const float* bih     = (const float*)d_in[5];
  const float* bhh     = (const float*)d_in[6];
  float* out = (float*)d_out;

  const size_t lds_floats = (size_t)S_ * XS_STRIDE + 2u * H_ * W_STRIDE +
                            3u * S_ + 4u * H_ + 32u;
  const size_t lds_bytes = lds_floats * sizeof(float);   // ~174 KB < 320 KB/WGP

  // Allow >64KB dynamic LDS (CDNA5 WGP has 320KB). Idempotent per call.
  (void)hipFuncSetAttribute((const void*)rnn_attn_scan,
                            hipFuncAttributeMaxDynamicSharedMemorySize,
                            (int)lds_bytes);

  rnn_attn_scan<<<B_, NTHREADS, lds_bytes, stream>>>(
      tokens, lengths, emb, Wih, Whh, bih, bhh, out);
}